// ResiduesDecodeVec_16363825397926
// MI455X (gfx1250) — compile-verified
//
#include <hip/hip_runtime.h>

typedef __attribute__((ext_vector_type(2))) float v2f;
typedef __attribute__((ext_vector_type(4))) float v4f;
typedef __attribute__((ext_vector_type(8))) float v8f;

#define B_      32
#define S_      512
#define VDIM_   256
#define MAXLEN_ 13
#define NCOL_   96          // B*3 columns of the per-position GEMM
#define LSTRIDE_ 260        // padded LDS row stride (floats): 260 % 64 == 4 -> conflict-free

// One workgroup per sequence position s.
// Phase 1: H = W_hid[seq[s]] (256x256) @ X (256x96)      via v_wmma_f32_16x16x4_f32
// Phase 2: Y = W_out[seq[s]] (13x256 pad 16) @ H (256x96) via v_wmma_f32_16x16x4_f32
__global__ __launch_bounds__(256)
void residues_gemm_kernel(const float* __restrict__ x_v,
                          const float* __restrict__ W_hid,
                          const float* __restrict__ W_out,
                          const int*  __restrict__ seq,
                          float* __restrict__ y_stage)
{
    extern __shared__ float smem[];       // 96 * 260 floats; holds X^T, then reused for H^T
    float* XT = smem;
    float* HT = smem;

    const int s     = blockIdx.x;
    const int tid   = threadIdx.x;
    const int lane  = tid & 31;
    const int wave  = tid >> 5;
    const int lhalf = lane >> 4;          // 0: K+0/K+1 half, 1: K+2/K+3 half (ISA A/B layout)
    const int l16   = lane & 15;

    const int letter = seq[s];

    // ---- stage X^T into LDS: XT[n][v] = x_v[b, s, v, c],  n = b*3 + c ----
    {
        const float* xs = x_v + (size_t)s * (VDIM_ * 3);
        for (int n = 0; n < NCOL_; ++n) {
            const int b = n / 3, c = n - 3 * b;
            const int v = tid;            // 256 threads cover the v dimension
            XT[n * LSTRIDE_ + v] = xs[((size_t)b * S_ * VDIM_ + v) * 3 + c];
        }
    }
    __syncthreads();

    // ---- phase 1: each wave computes rows [wave*32, wave*32+32) x all 96 cols ----
    const float* Wh = W_hid + (size_t)letter * VDIM_ * VDIM_;
    const int m0 = wave * 32;

    v8f acc[2][6];
#pragma unroll
    for (int mt = 0; mt < 2; ++mt)
#pragma unroll
        for (int nt = 0; nt < 6; ++nt)
            acc[mt][nt] = (v8f){0.f, 0.f, 0.f, 0.f, 0.f, 0.f, 0.f, 0.f};

    for (int k = 0; k < VDIM_; k += 4) {
        const int kk = k + 2 * lhalf;
        // A fragments: lane l16 = row m, lhalf selects K-pair (ISA 16x4 f32 A layout)
        v2f a0 = *(const v2f*)(Wh + (size_t)(m0 +      l16) * VDIM_ + kk);
        v2f a1 = *(const v2f*)(Wh + (size_t)(m0 + 16 + l16) * VDIM_ + kk);
#pragma unroll
        for (int nt = 0; nt < 6; ++nt) {
            // B fragment: lane l16 = column n, lhalf selects K-pair
            v2f b = *(const v2f*)(XT + (nt * 16 + l16) * LSTRIDE_ + kk);
            acc[0][nt] = __builtin_amdgcn_wmma_f32_16x16x4_f32(
                false, a0, false, b, (short)0, acc[0][nt], false, false);
            acc[1][nt] = __builtin_amdgcn_wmma_f32_16x16x4_f32(
                false, a1, false, b, (short)0, acc[1][nt], false, false);
        }
    }

    // all waves done reading XT before we overwrite the region with H^T
    __syncthreads();

    // D layout: lane holds column n0+l16, rows (8*lhalf + j), j=0..7 consecutive
#pragma unroll
    for (int mt = 0; mt < 2; ++mt) {
#pragma unroll
        for (int nt = 0; nt < 6; ++nt) {
            float* dst = HT + (nt * 16 + l16) * LSTRIDE_ + m0 + mt * 16 + 8 * lhalf;
            v8f A = acc[mt][nt];
            *(v4f*)(dst)     = (v4f){A[0], A[1], A[2], A[3]};
            *(v4f*)(dst + 4) = (v4f){A[4], A[5], A[6], A[7]};
        }
    }
    __syncthreads();

    // ---- phase 2: waves 0..5, each one 16x16 output tile (13 valid rows) ----
    if (wave < 6) {
        const float* Wo = W_out + (size_t)letter * MAXLEN_ * VDIM_;
        const int n0 = wave * 16;
        const bool mval = (l16 < MAXLEN_);
        const v2f zv2 = {0.f, 0.f};

        v8f acc2 = (v8f){0.f, 0.f, 0.f, 0.f, 0.f, 0.f, 0.f, 0.f};
        for (int k = 0; k < VDIM_; k += 4) {
            const int kk = k + 2 * lhalf;
            v2f a = mval ? *(const v2f*)(Wo + (size_t)l16 * VDIM_ + kk) : zv2;
            v2f b = *(const v2f*)(HT + (n0 + l16) * LSTRIDE_ + kk);
            acc2 = __builtin_amdgcn_wmma_f32_16x16x4_f32(
                false, a, false, b, (short)0, acc2, false, false);
        }

        // y_stage[(s*13 + m) * 96 + n]
        float* ys = y_stage + ((size_t)s * MAXLEN_) * NCOL_ + (n0 + l16);
#pragma unroll
        for (int j = 0; j < 8; ++j) {
            const int m = j + 8 * lhalf;
            if (m < MAXLEN_) ys[(size_t)m * NCOL_] = acc2[j];
        }
    }
}

// out[b, p, c] = y_stage[gather_idx[p] * 96 + b*3 + c]
__global__ __launch_bounds__(256)
void residues_gather_kernel(const float* __restrict__ y_stage,
                            const int* __restrict__ gidx,
                            float* __restrict__ out, int natoms)
{
    const int i = blockIdx.x * blockDim.x + threadIdx.x;
    const int tot = natoms * NCOL_;
    if (i >= tot) return;
    const int p = i / NCOL_;
    const int n = i - p * NCOL_;
    const int g = gidx[p];
    const int b = n / 3, c = n - 3 * b;
    out[((size_t)b * natoms + p) * 3 + c] = y_stage[(size_t)g * NCOL_ + n];
}

extern "C" void kernel_launch(void* const* d_in, const int* in_sizes, int n_in,
                              void* d_out, int out_size, void* d_ws, size_t ws_size,
                              hipStream_t stream) {
    const float* x_v  = (const float*)d_in[0];   // (32, 512, 256, 3) f32
    const float* Wh   = (const float*)d_in[1];   // (20, 256, 256)    f32
    const float* Wo   = (const float*)d_in[2];   // (20, 13, 256)     f32
    const int*   seq  = (const int*)  d_in[3];   // (512,)            i32
    const int*   gidx = (const int*)  d_in[4];   // (3745,)           i32
    float* out     = (float*)d_out;              // (32, 3745, 3)     f32
    float* y_stage = (float*)d_ws;               // 512*13*96 floats ≈ 2.6 MB

    const int natoms = in_sizes[4];
    const size_t lds_bytes = (size_t)NCOL_ * LSTRIDE_ * sizeof(float); // 99,840 B

    residues_gemm_kernel<<<S_, 256, lds_bytes, stream>>>(x_v, Wh, Wo, seq, y_stage);

    const int tot = natoms * NCOL_;
    residues_gather_kernel<<<(tot + 255) / 256, 256, 0, stream>>>(y_stage, gidx, out, natoms);
}